// CrossNetwork_40364102648060
// MI455X (gfx1250) — compile-verified
//
#include <hip/hip_runtime.h>

// CDNA5 / gfx1250: wave32, WMMA (not MFMA).
typedef __attribute__((ext_vector_type(2))) float v2f;
typedef __attribute__((ext_vector_type(8))) float v8f;

#define LAYERS 4
#define DIM 1024
#define ROWS_PER_WG 16
#define WAVES 16
#define THREADS (WAVES * 32)
#define COLS_PER_WAVE (DIM / WAVES)   // 64
#define CHUNKS (COLS_PER_WAVE / 4)    // 16 K-chunks of 4 for v_wmma_f32_16x16x4_f32

__launch_bounds__(THREADS)
__global__ void dcn_cross_kernel(const float* __restrict__ x,
                                 const float* __restrict__ w,
                                 const float* __restrict__ b,
                                 float* __restrict__ out)
{
    __shared__ __align__(16) float wlds[LAYERS * DIM];   // 16 KB weights
    __shared__ __align__(16) float blds[LAYERS * DIM];   // 16 KB bias
    __shared__ __align__(16) float pbuf[WAVES * 16];     // per-wave partial dots
    __shared__ __align__(16) float sred[16];             // reduced s per row

    const int tid  = threadIdx.x;
    const int wid  = tid >> 5;        // wave id 0..15
    const int lane = tid & 31;
    const int hh   = lane >> 4;       // 0: lanes 0-15 (K=0,1), 1: lanes 16-31 (K=2,3)
    const int m    = lane & 15;       // row within the 16-row tile (A-matrix M)
    const int colbase = wid * COLS_PER_WAVE;
    const int row  = blockIdx.x * ROWS_PER_WG + m;

    // ---- Stage weights + bias into LDS via async global->LDS DMA (ASYNCcnt). ----
    // Each lane: LDS byte offset in a VGPR, 64-bit global address in a VGPR pair.
    for (int i = tid; i < LAYERS * DIM / 4; i += THREADS) {
        unsigned lw = (unsigned)(unsigned long long)(uintptr_t)(&wlds[i * 4]);
        unsigned long long gw = (unsigned long long)(uintptr_t)(w + i * 4);
        asm volatile("global_load_async_to_lds_b128 %0, %1, off"
                     :: "v"(lw), "v"(gw) : "memory");
        unsigned lb = (unsigned)(unsigned long long)(uintptr_t)(&blds[i * 4]);
        unsigned long long gb = (unsigned long long)(uintptr_t)(b + i * 4);
        asm volatile("global_load_async_to_lds_b128 %0, %1, off"
                     :: "v"(lb), "v"(gb) : "memory");
    }

    // ---- Load x0 tile into registers in WMMA A-layout (overlaps the async DMA).
    // Lane holds, per chunk c, elements (colbase + 4c + 2*hh, +1) of its row:
    // exactly the 32-bit 16x4 A-matrix layout (VGPR0=K0/K2, VGPR1=K1/K3).
    const float* xrow = x + (size_t)row * DIM;
    float2 x0p[CHUNKS], xip[CHUNKS];
#pragma unroll
    for (int c = 0; c < CHUNKS; ++c) {
        int k0 = colbase + 4 * c + 2 * hh;
        x0p[c] = *(const float2*)(xrow + k0);
        xip[c] = x0p[c];
    }

    asm volatile("s_wait_asynccnt 0" ::: "memory");
    __syncthreads();   // all waves' async LDS writes visible workgroup-wide

#pragma unroll
    for (int layer = 0; layer < LAYERS; ++layer) {
        const float* wl = &wlds[layer * DIM];
        const float* bl = &blds[layer * DIM];

        // Partial dot of this wave's 16x64 xi slice with w via 16 accumulating
        // f32 WMMAs. B columns are all equal (broadcast w chunk), so every
        // D[m][n] equals the partial dot of row m.
        v8f acc = {0.f, 0.f, 0.f, 0.f, 0.f, 0.f, 0.f, 0.f};
#pragma unroll
        for (int c = 0; c < CHUNKS; ++c) {
            int k0 = colbase + 4 * c + 2 * hh;
            float2 wv = *(const float2*)(wl + k0);   // ds_load_b64, broadcast per half
            v2f av; av[0] = xip[c].x; av[1] = xip[c].y;
            v2f bv; bv[0] = wv.x;     bv[1] = wv.y;
            acc = __builtin_amdgcn_wmma_f32_16x16x4_f32(
                false, av, false, bv, (short)0, acc, false, false);
        }

        // D layout: VGPR i -> row i (lanes 0-15) / row i+8 (lanes 16-31).
        // Lane 0 publishes rows 0-7, lane 16 publishes rows 8-15.
        if (m == 0) {
            float* p = &pbuf[wid * 16 + hh * 8];
            p[0] = acc[0]; p[1] = acc[1]; p[2] = acc[2]; p[3] = acc[3];
            p[4] = acc[4]; p[5] = acc[5]; p[6] = acc[6]; p[7] = acc[7];
        }
        __syncthreads();

        // Reduce the 16 per-wave partials per row (16 lanes do 16 rows).
        if (tid < 16) {
            float s = 0.f;
#pragma unroll
            for (int ww = 0; ww < WAVES; ++ww) s += pbuf[ww * 16 + tid];
            sred[tid] = s;
        }
        __syncthreads();

        // Elementwise cross update: xi = x0 * s + (b + xi), in registers.
        float s = sred[m];
#pragma unroll
        for (int c = 0; c < CHUNKS; ++c) {
            int k0 = colbase + 4 * c + 2 * hh;
            float2 bb = *(const float2*)(bl + k0);
            xip[c].x = fmaf(x0p[c].x, s, bb.x + xip[c].x);
            xip[c].y = fmaf(x0p[c].y, s, bb.y + xip[c].y);
        }
    }

    // ---- Single streaming write of the final xi. ----
    float* orow = out + (size_t)row * DIM;
#pragma unroll
    for (int c = 0; c < CHUNKS; ++c) {
        int k0 = colbase + 4 * c + 2 * hh;
        *(float2*)(orow + k0) = xip[c];
    }
}

extern "C" void kernel_launch(void* const* d_in, const int* in_sizes, int n_in,
                              void* d_out, int out_size, void* d_ws, size_t ws_size,
                              hipStream_t stream) {
    const float* x = (const float*)d_in[0];           // [16384, 1024]
    const float* w = (const float*)d_in[1];           // [4, 1024]
    const float* b = (const float*)d_in[2];           // [4, 1024]
    float* out = (float*)d_out;                       // [16384, 1024]

    dim3 grid(16384 / ROWS_PER_WG);                   // 1024 workgroups
    dim3 block(THREADS);                              // 512 threads = 16 wave32
    hipLaunchKernelGGL(dcn_cross_kernel, grid, block, 0, stream, x, w, b, out);
}